// ConditionalRandomField_5755256177460
// MI455X (gfx1250) — compile-verified
//
#include <hip/hip_runtime.h>

// CRF log-likelihood on MI455X (gfx1250).
// B=128, S=1024, T=256. Forward scan rewritten as exp-domain GEMM per step:
//   alpha = c + log(v), max(v)=1;  step: w = (v^T x exp(T)) .* exp(emit);
//   c += log(max w); v = w / max w.   GEMM via V_WMMA_F32_16X16X4_F32.

#define B_  128
#define S_  1024
#define T_  256
#define MB  16      // batch tile per workgroup (M of the GEMM)
#define NWAVE 16    // waves per workgroup (512 threads, wave32)
#define VSTR 17     // padded row stride of transposed v  (bank-conflict free)
#define ESTR 260    // padded row stride of emit tiles    (bank-conflict free)

typedef float v2f __attribute__((ext_vector_type(2)));
typedef float v8f __attribute__((ext_vector_type(8)));

__launch_bounds__(512)
__global__ void crf_forward_scan(const float* __restrict__ logits,
                                 const unsigned char* __restrict__ mask,
                                 const float* __restrict__ trans,
                                 const float* __restrict__ start_t,
                                 const float* __restrict__ end_t,
                                 float* __restrict__ den)
{
    __shared__ float vT[T_ * VSTR];          // v transposed: [k][batch], stride 17
    __shared__ float emitb[2][MB * ESTR];    // double-buffered emit tiles
    __shared__ float pmax[MB * NWAVE];       // per-wave partial maxima
    __shared__ float mfin[MB];               // 1/max (>0) or -1 (masked: keep alpha)
    __shared__ float crow[MB];               // log-scale c per batch

    const int tid  = threadIdx.x;
    const int wave = tid >> 5;               // 0..15  (N-tile id)
    const int lane = tid & 31;
    const int half = lane >> 4;              // 0: lanes 0-15, 1: lanes 16-31
    const int l16  = lane & 15;
    const int col  = wave * 16 + l16;        // this lane's output column
    const int b0   = blockIdx.x * MB;

    // ---- Preload constant B fragments: E = exp(transitions), this wave's 16 cols.
    // A-layout analog (f32 16x16x4): VGPR0 holds K = 0 (lanes 0-15) / 2 (16-31),
    // VGPR1 holds K = 1 / 3.  -> Bfrag[kt] = { E[k0][col], E[k0+1][col] }.
    v2f Bfrag[64];
#pragma unroll
    for (int kt = 0; kt < 64; ++kt) {
        int k0 = kt * 4 + half * 2;
        Bfrag[kt].x = __expf(trans[k0 * T_ + col]);
        Bfrag[kt].y = __expf(trans[(k0 + 1) * T_ + col]);
    }

    // ---- Init: alpha0 = start_t + logits[:,0,:].  Wave w owns batch w.
    {
        const int bt = wave;
        float a[8];
        float mv = -1e30f;
#pragma unroll
        for (int i = 0; i < 8; ++i) {
            int k = lane + 32 * i;
            a[i] = start_t[k] + logits[(size_t)(b0 + bt) * S_ * T_ + k];
            mv = fmaxf(mv, a[i]);
        }
#pragma unroll
        for (int off = 1; off < 32; off <<= 1)
            mv = fmaxf(mv, __shfl_xor(mv, off, 32));
#pragma unroll
        for (int i = 0; i < 8; ++i) {
            int k = lane + 32 * i;
            vT[k * VSTR + bt] = __expf(a[i] - mv);
        }
        if (lane == 0) crow[bt] = mv;
    }

    // ---- Prefetch emit tile for step s=1 (thread t: batch=wave, 8 cols).
    {
        const int cl = (tid & 31) * 8;
        const float* src = logits + (size_t)(b0 + wave) * S_ * T_ + (size_t)1 * T_ + cl;
        float4 p0 = ((const float4*)src)[0];
        float4 p1 = ((const float4*)src)[1];
        *(float4*)&emitb[1][wave * ESTR + cl]     = p0;
        *(float4*)&emitb[1][wave * ESTR + cl + 4] = p1;
    }
    __syncthreads();

    // ==== Sequential scan over time ====
    for (int s = 1; s < S_; ++s) {
        const int cur = s & 1;

        // Software-pipelined prefetch of emit[s+1] into registers.
        float4 p0, p1;
        const bool havepf = (s + 1) < S_;
        if (havepf) {
            const float* src = logits + (size_t)(b0 + wave) * S_ * T_
                             + (size_t)(s + 1) * T_ + (size_t)((tid & 31) * 8);
            p0 = ((const float4*)src)[0];
            p1 = ((const float4*)src)[1];
        }

        // GEMM: q[m][col] = sum_k v[m][k] * E[k][col]   (M=16, N=16/wave, K=256)
        v8f acc0 = {0,0,0,0,0,0,0,0};
        v8f acc1 = {0,0,0,0,0,0,0,0};
        v8f acc2 = {0,0,0,0,0,0,0,0};
        v8f acc3 = {0,0,0,0,0,0,0,0};
#pragma unroll
        for (int kt = 0; kt < 64; ++kt) {
            int k0 = kt * 4 + half * 2;
            v2f a;
            a.x = vT[k0 * VSTR + l16];           // A: v[m=l16][k0]
            a.y = vT[(k0 + 1) * VSTR + l16];     // A: v[m=l16][k0+1]
            switch (kt & 3) {
            case 0: acc0 = __builtin_amdgcn_wmma_f32_16x16x4_f32(false, a, false, Bfrag[kt], (short)0, acc0, false, false); break;
            case 1: acc1 = __builtin_amdgcn_wmma_f32_16x16x4_f32(false, a, false, Bfrag[kt], (short)0, acc1, false, false); break;
            case 2: acc2 = __builtin_amdgcn_wmma_f32_16x16x4_f32(false, a, false, Bfrag[kt], (short)0, acc2, false, false); break;
            default:acc3 = __builtin_amdgcn_wmma_f32_16x16x4_f32(false, a, false, Bfrag[kt], (short)0, acc3, false, false); break;
            }
        }
        v8f q = (acc0 + acc1) + (acc2 + acc3);

        // Multiply exp(emit), local per-batch max over this wave's 16 cols.
        float wsv[8];
#pragma unroll
        for (int r = 0; r < 8; ++r) {
            const int batch = r + half * 8;      // C layout: VGPR r, half -> M
            float e = __expf(emitb[cur][batch * ESTR + col]);
            float w = q[r] * e;
            wsv[r] = w;
            float wm = w;
#pragma unroll
            for (int off = 1; off < 16; off <<= 1)
                wm = fmaxf(wm, __shfl_xor(wm, off, 32));
            if (l16 == r) pmax[batch * NWAVE + wave] = wm;
        }
        __syncthreads();

        // Wave0 finalizes per-batch max, applies mask, updates c, broadcasts 1/max.
        if (wave == 0 && lane < 16) {
            const int batch = lane;
            float M = pmax[batch * NWAVE + 0];
#pragma unroll
            for (int w2 = 1; w2 < NWAVE; ++w2)
                M = fmaxf(M, pmax[batch * NWAVE + w2]);
            unsigned char mk = mask[(size_t)(b0 + batch) * S_ + s];
            if (mk) { crow[batch] += __logf(M); mfin[batch] = 1.0f / M; }
            else    { mfin[batch] = -1.0f; }
        }
        __syncthreads();

        // Renormalize + write new v (transposed); masked batches keep old alpha.
#pragma unroll
        for (int r = 0; r < 8; ++r) {
            const int batch = r + half * 8;
            float rcp = mfin[batch];
            if (rcp > 0.0f) vT[col * VSTR + batch] = wsv[r] * rcp;
        }

        // Land the prefetched emit tile into the other buffer.
        if (havepf) {
            const int cl = (tid & 31) * 8;
            *(float4*)&emitb[cur ^ 1][wave * ESTR + cl]     = p0;
            *(float4*)&emitb[cur ^ 1][wave * ESTR + cl + 4] = p1;
        }
        __syncthreads();
    }

    // ---- Finalize: logZ[b] = c[b] + log( sum_j v[j][b] * exp(end_t[j]) )
    {
        const int bt = wave;
        float sacc = 0.0f;
#pragma unroll
        for (int i = 0; i < 8; ++i) {
            int k = lane + 32 * i;
            sacc += vT[k * VSTR + bt] * __expf(end_t[k]);
        }
#pragma unroll
        for (int off = 1; off < 32; off <<= 1)
            sacc += __shfl_xor(sacc, off, 32);
        if (lane == 0) den[b0 + bt] = crow[bt] + __logf(sacc);
    }
}

// ---- Numerator: score of the given tag path (gather + deterministic reduce).
__global__ void crf_joint(const float* __restrict__ logits,
                          const int* __restrict__ tags,
                          const unsigned char* __restrict__ mask,
                          const float* __restrict__ trans,
                          const float* __restrict__ start_t,
                          const float* __restrict__ end_t,
                          float* __restrict__ num)
{
    __shared__ float red[256];
    __shared__ int   redi[256];
    const int b = blockIdx.x, t = threadIdx.x;
    float part = 0.0f;
    int   cnt  = 0;
    for (int s = t; s < S_; s += 256) {
        int tg = tags[b * S_ + s];
        int mk = mask[(size_t)b * S_ + s] ? 1 : 0;
        cnt += mk;
        float mf = (float)mk;
        if (s < S_ - 1) part += logits[((size_t)b * S_ + s) * T_ + tg] * mf;
        if (s >= 1)     part += trans[tags[b * S_ + s - 1] * T_ + tg] * mf;
    }
    red[t] = part; redi[t] = cnt;
    __syncthreads();
    for (int off = 128; off > 0; off >>= 1) {
        if (t < off) { red[t] += red[t + off]; redi[t] += redi[t + off]; }
        __syncthreads();
    }
    if (t == 0) {
        int   last = redi[0] - 1;
        int   lt   = tags[b * S_ + last];
        float mfl  = mask[(size_t)b * S_ + (S_ - 1)] ? 1.0f : 0.0f;
        num[b] = red[0] + start_t[tags[b * S_]] + end_t[lt]
               + logits[((size_t)b * S_ + (S_ - 1)) * T_ + lt] * mfl;
    }
}

// ---- Final deterministic reduction: out = sum_b (num[b] - den[b]).
__global__ void crf_final(const float* __restrict__ num,
                          const float* __restrict__ den,
                          float* __restrict__ out)
{
    __shared__ float red[128];
    const int t = threadIdx.x;
    red[t] = num[t] - den[t];
    __syncthreads();
    for (int off = 64; off > 0; off >>= 1) {
        if (t < off) red[t] += red[t + off];
        __syncthreads();
    }
    if (t == 0) out[0] = red[0];
}

extern "C" void kernel_launch(void* const* d_in, const int* in_sizes, int n_in,
                              void* d_out, int out_size, void* d_ws, size_t ws_size,
                              hipStream_t stream)
{
    const float*         logits = (const float*)d_in[0];
    const int*           tags   = (const int*)d_in[1];
    const unsigned char* mask   = (const unsigned char*)d_in[2];
    const float*         trans  = (const float*)d_in[3];
    const float*         st     = (const float*)d_in[4];
    const float*         et     = (const float*)d_in[5];

    float* ws  = (float*)d_ws;
    float* den = ws;          // [128]
    float* num = ws + 128;    // [128]

    crf_forward_scan<<<dim3(B_ / MB), dim3(512), 0, stream>>>(logits, mask, trans, st, et, den);
    crf_joint<<<dim3(B_), dim3(256), 0, stream>>>(logits, tags, mask, trans, st, et, num);
    crf_final<<<1, 128, 0, stream>>>(num, den, (float*)d_out);
}